// AugmentationModule_85409719648781
// MI455X (gfx1250) — compile-verified
//
#include <hip/hip_runtime.h>
#include <math.h>

typedef __attribute__((ext_vector_type(2))) float v2f;
typedef __attribute__((ext_vector_type(8))) float v8f;

#define N_NODES   32768
#define NUM_G     8
#define M_NODES   4096          // nodes per graph
#define KNN       50
#define NUM_BINS  5
#define TILE      16            // WMMA tile edge
#define ROWS      32            // rows per wave (2 WMMA row-tiles)
#define LDSW      20            // LDS row stride (floats): 16B-aligned rows
#define NKE       ((long)N_NODES * KNN)   // 1,638,400 (half-edge count)

// one wave (32 threads) per 32-node row block; every lane owns one row
__launch_bounds__(32)
__global__ void knn_rdf_kernel(const float* __restrict__ pos,
                               float* __restrict__ out)
{
    const int blk      = blockIdx.x;            // 0 .. 1023
    const int lane     = threadIdx.x;           // 0 .. 31
    const int row_base = blk * ROWS;            // global node id of row 0
    const int g        = row_base / M_NODES;    // graph id (ROWS divides M_NODES)
    const int cg_base  = g * M_NODES;           // first node of this graph

    __shared__ __align__(16) float lds_tile[ROWS][LDSW];

    // ---- A fragments (16x4 f32): row i = (x, y, z, |p|^2), constant over loop
    // lanes 0-15 hold K=0,1 (x,y); lanes 16-31 hold K=2,3 (z,|p|^2)
    const int rm = lane & 15;
    const int r0 = row_base + rm;               // rows 0..15 of block
    const int r1 = row_base + 16 + rm;          // rows 16..31 of block
    const float x0 = pos[3 * r0 + 0], y0 = pos[3 * r0 + 1], z0 = pos[3 * r0 + 2];
    const float x1 = pos[3 * r1 + 0], y1 = pos[3 * r1 + 1], z1 = pos[3 * r1 + 2];
    const float n0 = x0 * x0 + y0 * y0 + z0 * z0;
    const float n1 = x1 * x1 + y1 * y1 + z1 * z1;
    v2f a0, a1;
    if (lane < 16) { a0.x = x0; a0.y = y0; a1.x = x1; a1.y = y1; }
    else           { a0.x = z0; a0.y = n0; a1.x = z1; a1.y = n1; }

    const int grow = row_base + lane;           // the row this lane owns

    // per-row top-K max-heap (d2, index); hmax caches hd[0] in a register
    float hd[KNN];
    int   hi[KNN];
    #pragma unroll
    for (int k = 0; k < KNN; ++k) { hd[k] = 3.0e38f; hi[k] = 0; }
    float hmax = 3.0e38f;

    #pragma unroll 1
    for (int ct = 0; ct < M_NODES / TILE; ++ct) {
        const int col_base = cg_base + ct * TILE;
        const int gcol     = col_base + rm;
        const float cx = pos[3 * gcol + 0];
        const float cy = pos[3 * gcol + 1];
        const float cz = pos[3 * gcol + 2];
        const float cn2 = cx * cx + cy * cy + cz * cz;

        // prefetch next column tile's positions (global_prefetch_b8)
        if (ct + 1 < M_NODES / TILE)
            __builtin_prefetch(&pos[3 * (gcol + TILE)], 0, 1);

        // ---- B fragment (4x16 f32): col j = (-2x, -2y, -2z, 1)^T
        v2f bf;
        if (lane < 16) { bf.x = -2.0f * cx; bf.y = -2.0f * cy; }
        else           { bf.x = -2.0f * cz; bf.y = 1.0f; }

        // ---- C fragment: broadcast |p_j|^2 of this lane's column
        v8f cf;
        #pragma unroll
        for (int r = 0; r < 8; ++r) cf[r] = cn2;

        // D[i][j] = -2 pi.pj + |pi|^2 + |pj|^2 = squared distance
        v8f d0 = __builtin_amdgcn_wmma_f32_16x16x4_f32(
            false, a0, false, bf, (short)0, cf, false, false);
        v8f d1 = __builtin_amdgcn_wmma_f32_16x16x4_f32(
            false, a1, false, bf, (short)0, cf, false, false);

        // spill: VGPR r -> row r (lanes 0-15) / row r+8 (lanes 16-31), col = rm
        const int mrow = (lane < 16) ? 0 : 8;
        #pragma unroll
        for (int r = 0; r < 8; ++r) {
            lds_tile[mrow + r][rm]      = d0[r];
            lds_tile[16 + mrow + r][rm] = d1[r];
        }
        asm volatile("s_wait_dscnt 0" ::: "memory");

        // bulk-load this lane's row (4x ds_load_b128) and min-reduce it
        const float4 q0 = *(const float4*)&lds_tile[lane][0];
        const float4 q1 = *(const float4*)&lds_tile[lane][4];
        const float4 q2 = *(const float4*)&lds_tile[lane][8];
        const float4 q3 = *(const float4*)&lds_tile[lane][12];
        const float mx = fminf(fminf(q0.x, q1.x), fminf(q2.x, q3.x));
        const float my = fminf(fminf(q0.y, q1.y), fminf(q2.y, q3.y));
        const float mz = fminf(fminf(q0.z, q1.z), fminf(q2.z, q3.z));
        const float mw = fminf(fminf(q0.w, q1.w), fminf(q2.w, q3.w));
        const float tmin = fminf(fminf(mx, my), fminf(mz, mw));

        // rare path: this tile holds at least one new top-50 candidate
        // (the self column d2~0 also triggers once; it is skipped below)
        if (tmin < hmax) {
            const int selfc = grow - col_base;
            #pragma unroll 1
            for (int c = 0; c < TILE; ++c) {
                if (c == selfc) continue;            // knn_graph loop=False
                const float d2 = lds_tile[lane][c];
                if (d2 < hmax) {
                    hd[0] = d2; hi[0] = col_base + c; // replace max, sift down
                    int p = 0;
                    for (;;) {
                        const int l = 2 * p + 1, r2 = 2 * p + 2;
                        int big = p;
                        if (l  < KNN && hd[l]  > hd[big]) big = l;
                        if (r2 < KNN && hd[r2] > hd[big]) big = r2;
                        if (big == p) break;
                        float td = hd[p]; hd[p] = hd[big]; hd[big] = td;
                        int   ti = hi[p]; hi[p] = hi[big]; hi[big] = ti;
                        p = big;
                    }
                    hmax = hd[0];                     // refresh threshold
                }
            }
        }
    }

    // heapsort -> ascending d2 (matches top_k neighbor ordering)
    for (int sz = KNN - 1; sz > 0; --sz) {
        float td = hd[0]; hd[0] = hd[sz]; hd[sz] = td;
        int   ti = hi[0]; hi[0] = hi[sz]; hi[sz] = ti;
        int p = 0;
        for (;;) {
            const int l = 2 * p + 1, r2 = 2 * p + 2;
            int big = p;
            if (l  < sz && hd[l]  > hd[big]) big = l;
            if (r2 < sz && hd[r2] > hd[big]) big = r2;
            if (big == p) break;
            float t2 = hd[p]; hd[p] = hd[big]; hd[big] = t2;
            int   t3 = hi[p]; hi[p] = hi[big]; hi[big] = t3;
            p = big;
        }
    }

    // emit edges: edge_index = [concat(src,dst), concat(dst,src)]
    const int i = grow;
    const float px = pos[3 * i + 0], py = pos[3 * i + 1], pz = pos[3 * i + 2];
    const float width  = 10.0f / (NUM_BINS - 1);     // 2.5
    const float inv2w2 = 1.0f / (2.0f * width * width);
    #pragma unroll 1
    for (int k = 0; k < KNN; ++k) {
        const int  nb = hi[k];
        const float dx = px - pos[3 * nb + 0];
        const float dy = py - pos[3 * nb + 1];
        const float dz = pz - pos[3 * nb + 2];
        const float dist = sqrtf(dx * dx + dy * dy + dz * dz);
        const long e = (long)i * KNN + k;

        out[e]           = (float)nb;   // edge_index[0], first half  (src)
        out[NKE + e]     = (float)i;    // edge_index[0], second half (dst)
        out[2 * NKE + e] = (float)i;    // edge_index[1], first half  (dst)
        out[3 * NKE + e] = (float)nb;   // edge_index[1], second half (src)
        out[4 * NKE + e] = dist;        // dist, first half
        out[5 * NKE + e] = dist;        // dist, second half

        #pragma unroll
        for (int b = 0; b < NUM_BINS; ++b) {
            const float t = dist - width * (float)b;
            const float a = __expf(-(t * t) * inv2w2);
            out[6 * NKE + e * NUM_BINS + b]         = a;   // attr, first half
            out[6 * NKE + (NKE + e) * NUM_BINS + b] = a;   // attr, second half
        }
    }
}

extern "C" void kernel_launch(void* const* d_in, const int* in_sizes, int n_in,
                              void* d_out, int out_size, void* d_ws, size_t ws_size,
                              hipStream_t stream) {
    (void)in_sizes; (void)n_in; (void)d_ws; (void)ws_size; (void)out_size;
    const float* pos = (const float*)d_in[0];
    float* out = (float*)d_out;
    dim3 grid(N_NODES / ROWS);   // 1024 blocks of 32 nodes
    dim3 block(32);              // one wave32 per block
    hipLaunchKernelGGL(knn_rdf_kernel, grid, block, 0, stream, pos, out);
}